// TDa_78675210928333
// MI455X (gfx1250) — compile-verified
//
#include <hip/hip_runtime.h>
#include <hip/hip_bf16.h>
#include <math.h>

#define TT   12
#define NN   10000
#define EE   160000
#define PP   30000
#define DIN  128
#define HH   64
#define NG   24          // 2*T graphs
#define LDSS 132         // padded LDS row stride (floats), keeps 8B align, spreads banks

typedef __attribute__((ext_vector_type(2))) float v2f;
typedef __attribute__((ext_vector_type(8))) float v8f;

// ---------------------------------------------------------------- utilities
__device__ __forceinline__ float wave_sum(float v) {
#pragma unroll
    for (int o = 16; o > 0; o >>= 1) v += __shfl_xor(v, o, 32);
    return v;
}

// ---------------------------------------------------------------- zero msum/deg
__global__ __launch_bounds__(256) void zero_kernel(float* msum, float* deg) {
    int i = blockIdx.x * 256 + threadIdx.x;
    if (i < NN * DIN) msum[i] = 0.0f;
    if (i < NN)       deg[i]  = 0.0f;
}

// ---------------------------------------------------------------- one-time weight transpose [DIN,H] -> [H,DIN]
__global__ __launch_bounds__(256) void transpose_w_kernel(const float* __restrict__ w_self,
                                                          const float* __restrict__ w_neigh,
                                                          float* __restrict__ wsT,
                                                          float* __restrict__ wnT) {
    int i = blockIdx.x * 256 + threadIdx.x;       // over DIN*HH = 8192
    if (i < DIN * HH) {
        int k = i / HH, c = i % HH;
        wsT[c * DIN + k] = w_self[i];
        wnT[c * DIN + k] = w_neigh[i];
    }
}

// ---------------------------------------------------------------- edge scatter (segment_sum), wave per edge
__global__ __launch_bounds__(256) void scatter_kernel(const float* __restrict__ xg,
                                                      const int* __restrict__ src,
                                                      const int* __restrict__ dst,
                                                      float* __restrict__ msum,
                                                      float* __restrict__ deg) {
    const int lane = threadIdx.x & 31;
    const int e    = blockIdx.x * 8 + (threadIdx.x >> 5);
    const int s    = src[e];
    const int d    = dst[e];
    float4 xv = ((const float4*)(xg + (long)s * DIN))[lane];   // coalesced 512B/row
    float* mb = msum + (long)d * DIN + lane * 4;
    atomicAdd(mb + 0, xv.x);
    atomicAdd(mb + 1, xv.y);
    atomicAdd(mb + 2, xv.z);
    atomicAdd(mb + 3, xv.w);
    if (lane == 0) atomicAdd(&deg[d], 1.0f);
}

// ---------------------------------------------------------------- SAGE GEMM via V_WMMA_F32_16X16X4_F32
// h[16,64] = x_tile @ w_self + (msum_tile/deg) @ w_neigh + b, then fused row L2-normalize.
// block = 128 threads = 4 waves; wave w owns H-columns [16w, 16w+16)
__global__ __launch_bounds__(128) void sage_wmma_kernel(const float* __restrict__ xg,
                                                        const float* __restrict__ msum,
                                                        const float* __restrict__ deg,
                                                        const float* __restrict__ wsT,  // [H,DIN]
                                                        const float* __restrict__ wnT,  // [H,DIN]
                                                        const float* __restrict__ b_sage,
                                                        float* __restrict__ hg) {
    __shared__ float xs[16 * LDSS];
    __shared__ float ms[16 * LDSS];
    __shared__ float sq[16];
    const int tid      = threadIdx.x;
    const int row_base = blockIdx.x * 16;

    if (tid < 16) sq[tid] = 0.0f;

    // staged load of x tile and mean tile (mean division fused here)
#pragma unroll
    for (int i = tid; i < 16 * 32; i += 128) {
        int r  = i >> 5;
        int c4 = i & 31;
        float rdeg = 1.0f / fmaxf(deg[row_base + r], 1.0f);
        float4 xv = ((const float4*)(xg   + (long)(row_base + r) * DIN))[c4];
        float4 mv = ((const float4*)(msum + (long)(row_base + r) * DIN))[c4];
        float* xd = &xs[r * LDSS + c4 * 4];
        float* md = &ms[r * LDSS + c4 * 4];
        xd[0] = xv.x;        xd[1] = xv.y;        xd[2] = xv.z;        xd[3] = xv.w;
        md[0] = mv.x * rdeg; md[1] = mv.y * rdeg; md[2] = mv.z * rdeg; md[3] = mv.w * rdeg;
    }
    __syncthreads();

    const int lane = tid & 31;
    const int wv   = tid >> 5;        // H tile 0..3
    const int half = lane >> 4;       // 0: K=k0,k0+1   1: K=k0+2,k0+3 (ISA 16x4 layout)
    const int lr   = lane & 15;       // M (for A) / N (for B)
    const int col  = wv * 16 + lr;    // output H column of this lane

    v8f c = {};
#pragma unroll
    for (int kk = 0; kk < 32; ++kk) {
        const int k0 = kk * 4 + half * 2;
        v2f a_s = *(const v2f*)&xs[lr * LDSS + k0];        // ds_load_b64
        v2f a_n = *(const v2f*)&ms[lr * LDSS + k0];
        v2f b_s = *(const v2f*)&wsT[(long)col * DIN + k0]; // global_load_b64
        v2f b_n = *(const v2f*)&wnT[(long)col * DIN + k0];
        c = __builtin_amdgcn_wmma_f32_16x16x4_f32(false, a_s, false, b_s,
                                                  (short)0, c, false, false);
        c = __builtin_amdgcn_wmma_f32_16x16x4_f32(false, a_n, false, b_n,
                                                  (short)0, c, false, false);
    }

    // epilogue: bias + fused L2 row-normalize
    const float bias = b_sage[col];
    float val[8];
#pragma unroll
    for (int v = 0; v < 8; ++v) {
        val[v] = c[v] + bias;
        float s2 = val[v] * val[v];
#pragma unroll
        for (int o = 1; o < 16; o <<= 1) s2 += __shfl_xor(s2, o, 32); // 16-lane group reduce
        if (lr == 0) atomicAdd(&sq[half * 8 + v], s2);                // ds_add_f32
    }
    __syncthreads();

#pragma unroll
    for (int v = 0; v < 8; ++v) {
        int row = half * 8 + v;                    // C/D layout: VGPR v -> M = v + 8*half
        float scale = 1.0f / fmaxf(sqrtf(sq[row]), 1e-12f);
        hg[(long)(row_base + row) * HH + col] = val[v] * scale;
    }
}

// ---------------------------------------------------------------- fused temporal window attention
// one wave per pair p; lane holds channels c = lane + 32q, q=0..3 (2H = 128)
__global__ __launch_bounds__(256) void pair_kernel(const float* __restrict__ hbuf,
                                                   const int* __restrict__ idx,
                                                   const float* __restrict__ w_ff2,
                                                   const float* __restrict__ b_ff2,
                                                   const float* __restrict__ w_ff1,
                                                   const float* __restrict__ b_ff1,
                                                   const float* __restrict__ w_out,
                                                   float* __restrict__ out) {
    const int lane = threadIdx.x & 31;
    const int p    = blockIdx.x * 8 + (threadIdx.x >> 5);
    if (p >= PP) return;

    float tp[TT][4];
#pragma unroll
    for (int t = 0; t < TT; ++t) {
        int i0 = idx[t * PP + p];                 // idx[0][t][p]
        int i1 = idx[(TT + t) * PP + p];          // idx[1][t][p]
        const float* h0 = hbuf + ((long)t * NN + i0) * HH;          // graph g = t
        const float* h1 = hbuf + ((long)(TT + t) * NN + i1) * HH;   // graph g = 12+t
        tp[t][0] = h0[lane];
        tp[t][1] = h0[lane + 32];
        tp[t][2] = h1[lane];
        tp[t][3] = h1[lane + 32];
    }

    float wf2[4], wo[4];
#pragma unroll
    for (int q = 0; q < 4; ++q) {
        wf2[q] = w_ff2[lane + 32 * q];
        wo[q]  = w_out[lane + 32 * q];
    }
    const float bf2 = b_ff2[0];
    const float bf1 = b_ff1[0];
    float wt[TT];
#pragma unroll
    for (int t = 0; t < TT; ++t) wt[t] = w_ff1[t];

    // s[k] = sum_t relu(diffs[k,t] . w_ff2 + b_ff2) * w_ff1[t] + b_ff1
    float s[3];
#pragma unroll
    for (int k = 0; k < 3; ++k) {
        float acc = 0.0f;
#pragma unroll
        for (int t = 0; t < TT; ++t) {
            float partial = 0.0f;
#pragma unroll
            for (int q = 0; q < 4; ++q) {
                float d = tp[t][q] - ((t > k) ? tp[t - k - 1][q] : 0.0f);
                partial += d * wf2[q];
            }
            float dot = wave_sum(partial) + bf2;
            acc += fmaxf(dot, 0.0f) * wt[t];
        }
        s[k] = acc + bf1;
    }

    // softmax over window axis (3 entries, identical across lanes)
    float m  = fmaxf(s[0], fmaxf(s[1], s[2]));
    float e0 = expf(s[0] - m), e1 = expf(s[1] - m), e2 = expf(s[2] - m);
    float inv = 1.0f / (e0 + e1 + e2);
    float swk[3] = {e0 * inv, e1 * inv, e2 * inv};

#pragma unroll
    for (int t = 0; t < TT; ++t) {
        float po = 0.0f;
#pragma unroll
        for (int q = 0; q < 4; ++q) {
            float att = 0.0f;
#pragma unroll
            for (int k = 0; k < 3; ++k) {
                float d = tp[t][q] - ((t > k) ? tp[t - k - 1][q] : 0.0f);
                att += swk[k] / (1.0f + expf(-d));    // sw * sigmoid(diff)
            }
            po += tp[t][q] * att * wo[q];
        }
        float o = wave_sum(po);
        if (lane == 0) out[t * PP + p] = o;
    }
}

// ---------------------------------------------------------------- launch
extern "C" void kernel_launch(void* const* d_in, const int* in_sizes, int n_in,
                              void* d_out, int out_size, void* d_ws, size_t ws_size,
                              hipStream_t stream) {
    const float* x       = (const float*)d_in[0];    // [2,T,N,DIN]
    const int*   src     = (const int*)d_in[1];      // [2,T,E]
    const int*   dst     = (const int*)d_in[2];      // [2,T,E]
    const int*   idx     = (const int*)d_in[3];      // [2,T,P]
    const float* w_self  = (const float*)d_in[4];    // [DIN,H]
    const float* w_neigh = (const float*)d_in[5];    // [DIN,H]
    const float* b_sage  = (const float*)d_in[6];    // [H]
    const float* w_ff2   = (const float*)d_in[7];    // [2H]
    const float* b_ff2   = (const float*)d_in[8];    // scalar
    const float* w_ff1   = (const float*)d_in[9];    // [T]
    const float* b_ff1   = (const float*)d_in[10];   // scalar
    const float* w_out   = (const float*)d_in[11];   // [2H,1]
    float*       out     = (float*)d_out;            // [T,P]

    // workspace carve: msum[N,128] | deg[N] | hbuf[24,N,64] | wsT[64,128] | wnT[64,128]
    float* msum = (float*)d_ws;
    float* deg  = msum + (long)NN * DIN;
    float* hbuf = deg + NN;
    float* wsT  = hbuf + (long)NG * NN * HH;
    float* wnT  = wsT + DIN * HH;

    transpose_w_kernel<<<(DIN * HH + 255) / 256, 256, 0, stream>>>(w_self, w_neigh, wsT, wnT);

    const int zgrid = (NN * DIN + 255) / 256;
    for (int g = 0; g < NG; ++g) {
        const float* xg = x + (long)g * NN * DIN;
        zero_kernel<<<zgrid, 256, 0, stream>>>(msum, deg);
        scatter_kernel<<<EE / 8, 256, 0, stream>>>(xg, src + (long)g * EE,
                                                   dst + (long)g * EE, msum, deg);
        sage_wmma_kernel<<<NN / 16, 128, 0, stream>>>(xg, msum, deg, wsT, wnT, b_sage,
                                                      hbuf + (long)g * NN * HH);
    }
    pair_kernel<<<PP / 8, 256, 0, stream>>>(hbuf, idx, w_ff2, b_ff2, w_ff1, b_ff1,
                                            w_out, out);
}